// ChebNet_34565896798961
// MI455X (gfx1250) — compile-verified
//
#include <hip/hip_runtime.h>

typedef float v2f __attribute__((ext_vector_type(2)));
typedef float v8f __attribute__((ext_vector_type(8)));

// ---------------------------------------------------------------------------
// Elementwise helpers
// ---------------------------------------------------------------------------
__global__ void zero4_kernel(float4* __restrict__ p, long long n4) {
    long long i = (long long)blockIdx.x * blockDim.x + threadIdx.x;
    if (i < n4) p[i] = make_float4(0.f, 0.f, 0.f, 0.f);
}

__global__ void negcopy4_kernel(float4* __restrict__ d, const float4* __restrict__ s,
                                long long n4) {
    long long i = (long long)blockIdx.x * blockDim.x + threadIdx.x;
    if (i < n4) {
        float4 v = s[i];
        d[i] = make_float4(-v.x, -v.y, -v.z, -v.w);
    }
}

// ---------------------------------------------------------------------------
// Laplacian normalization: deg = segsum(w, row); dinv = rsqrt(deg);
// nw[e] = -(dinv[row] * w * dinv[col])
// ---------------------------------------------------------------------------
__global__ void deg_accum_kernel(const int* __restrict__ ei, const float* __restrict__ ew,
                                 float* __restrict__ deg, int E) {
    int e = blockIdx.x * blockDim.x + threadIdx.x;
    if (e < E) atomicAdd(&deg[ei[e]], ew[e]);
}

__global__ void make_dinv_kernel(float* __restrict__ deg, int n) {
    int i = blockIdx.x * blockDim.x + threadIdx.x;
    if (i < n) {
        float d = deg[i];
        deg[i] = (d > 0.f) ? rsqrtf(d) : 0.f;
    }
}

__global__ void make_norm_kernel(const int* __restrict__ ei, const float* __restrict__ ew,
                                 const float* __restrict__ dinv, float* __restrict__ nw,
                                 int E) {
    int e = blockIdx.x * blockDim.x + threadIdx.x;
    if (e < E) {
        int r = ei[e];
        int c = ei[E + e];
        nw[e] = -(dinv[r] * ew[e] * dinv[c]);
    }
}

// ---------------------------------------------------------------------------
// SpMM scatter: vout[row] += scale * nw[e] * vin[col]   (one wave per edge)
// C channels, each lane owns C/32 channels as float4 chunks.
// ---------------------------------------------------------------------------
template <int C>
__global__ __launch_bounds__(256) void spmm_scatter_kernel(
    const int* __restrict__ ei, const float* __restrict__ nw,
    const float* __restrict__ vin, float* __restrict__ vout, float scale, int E) {
    long long gid = (long long)blockIdx.x * blockDim.x + threadIdx.x;
    int e = (int)(gid >> 5);
    int lane = (int)(gid & 31);
    if (e >= E) return;
    int r = ei[e];
    int c = ei[E + e];
    float w = nw[e] * scale;
    const float* src = vin + (size_t)c * C;
    float* dst = vout + (size_t)r * C;
#pragma unroll
    for (int v = 0; v < C / 128; ++v) {
        int cc = v * 128 + lane * 4;
        float4 x4 = *(const float4*)(src + cc);
        atomicAdd(dst + cc + 0, w * x4.x);
        atomicAdd(dst + cc + 1, w * x4.y);
        atomicAdd(dst + cc + 2, w * x4.z);
        atomicAdd(dst + cc + 3, w * x4.w);
    }
}

// ---------------------------------------------------------------------------
// Fused Chebyshev GEMM: out = act( A0@W[0] + A1@W[1] + A2@W[2] + bias )
// W flat: [3][CIN][COUT].  One 16x16 D-tile per wave via V_WMMA_F32_16X16X4_F32,
// 8 waves/block -> 128x16 tile, B panel ([3*CIN]x16) staged in LDS.
//
// Fragment layouts per CDNA5 ISA 7.12.2 (wave32):
//   A 16x4 f32:  M = lane&15 (both halves); VGPR0 = K+2*(lane>>4), VGPR1 = K+1+2*(lane>>4)
//   B 4x16 f32:  N = lane&15; VGPR0 = row K+2*(lane>>4), VGPR1 = row K+1+2*(lane>>4)
//   C/D 16x16:   N = lane&15; VGPR r holds M = r + 8*(lane>>4)
// ---------------------------------------------------------------------------
template <int CIN, int COUT>
__global__ __launch_bounds__(256) void cheb_gemm_kernel(
    const float* __restrict__ A0, const float* __restrict__ A1,
    const float* __restrict__ A2, const float* __restrict__ W,
    const float* __restrict__ bias, float* __restrict__ out, int nrows, int dorelu) {
    __shared__ float Bs[3 * CIN * 16];

    const int n0 = blockIdx.y * 16;
    // Cooperative stage of the B panel (rows = concatenated K of the 3 terms).
    for (int idx = threadIdx.x; idx < 3 * CIN * 16; idx += 256) {
        int krow = idx >> 4;
        int n = idx & 15;
        Bs[idx] = W[(size_t)krow * COUT + n0 + n];
    }
    __syncthreads();

    const int wave = threadIdx.x >> 5;
    const int lane = threadIdx.x & 31;
    const int m0 = (blockIdx.x * 8 + wave) * 16;
    if (m0 >= nrows) return;

    const int mh = lane >> 4;  // half selector (K-pair for A/B, M+8 for C/D)
    const int ml = lane & 15;

    v8f acc = {0.f, 0.f, 0.f, 0.f, 0.f, 0.f, 0.f, 0.f};
    const float* Asegs[3] = {A0, A1, A2};

#pragma unroll
    for (int kk = 0; kk < 3; ++kk) {
        const float* arow = Asegs[kk] + (size_t)(m0 + ml) * CIN + 2 * mh;
#pragma unroll 8
        for (int k = 0; k < CIN; k += 4) {
            float2 at = *(const float2*)(arow + k);
            v2f a;
            a[0] = at.x;
            a[1] = at.y;
            int krow = kk * CIN + k + 2 * mh;
            v2f b;
            b[0] = Bs[krow * 16 + ml];
            b[1] = Bs[(krow + 1) * 16 + ml];
            acc = __builtin_amdgcn_wmma_f32_16x16x4_f32(
                /*neg_a=*/false, a, /*neg_b=*/false, b,
                /*c_mod=*/(short)0, acc, /*reuse_a=*/false, /*reuse_b=*/false);
        }
    }

    float bv = bias[n0 + ml];
#pragma unroll
    for (int r = 0; r < 8; ++r) {
        int row = m0 + r + 8 * mh;
        float v = acc[r] + bv;
        if (dorelu) v = fmaxf(v, 0.f);
        out[(size_t)row * COUT + n0 + ml] = v;
    }
}

// ---------------------------------------------------------------------------
// Host-side orchestration
// ---------------------------------------------------------------------------
extern "C" void kernel_launch(void* const* d_in, const int* in_sizes, int n_in,
                              void* d_out, int out_size, void* d_ws, size_t ws_size,
                              hipStream_t stream) {
    (void)in_sizes; (void)n_in; (void)out_size; (void)ws_size;
    const float* x  = (const float*)d_in[0];
    const int*   ei = (const int*)d_in[1];      // [2][E] row-major
    const float* ew = (const float*)d_in[2];
    const float* W1 = (const float*)d_in[3];    // [3][128][256]
    const float* b1 = (const float*)d_in[4];
    const float* W2 = (const float*)d_in[5];    // [3][256][64]
    const float* b2 = (const float*)d_in[6];
    float* out = (float*)d_out;                 // [N][64]

    const int N = 100000, E = 3200000, CI = 128, CH = 256, CO = 64;

    float* ws = (float*)d_ws;
    size_t o = 0;
    float* nw   = ws + o; o += (size_t)E;
    float* deg  = ws + o; o += (size_t)N;
    float* tx1a = ws + o; o += (size_t)N * CI;
    float* tx2a = ws + o; o += (size_t)N * CI;
    float* h    = ws + o; o += (size_t)N * CH;
    float* tx1b = tx1a;  // reuse [tx1a|tx2a] region (exactly N*256 floats)
    float* tx2b = ws + o; o += (size_t)N * CH;

    // ---- Laplacian normalization ----
    zero4_kernel<<<(N / 4 + 255) / 256, 256, 0, stream>>>((float4*)deg, N / 4);
    deg_accum_kernel<<<(E + 255) / 256, 256, 0, stream>>>(ei, ew, deg, E);
    make_dinv_kernel<<<(N + 255) / 256, 256, 0, stream>>>(deg, N);
    make_norm_kernel<<<(E + 255) / 256, 256, 0, stream>>>(ei, ew, deg, nw, E);

    const long long spmm_threads = (long long)E * 32;
    const int spmm_blocks = (int)((spmm_threads + 255) / 256);

    // ---- Layer 1 propagation: Tx1 = L x ; Tx2 = 2 L Tx1 - x ----
    long long n1 = (long long)N * CI;
    zero4_kernel<<<(int)(n1 / 4 + 255) / 256, 256, 0, stream>>>((float4*)tx1a, n1 / 4);
    spmm_scatter_kernel<128><<<spmm_blocks, 256, 0, stream>>>(ei, nw, x, tx1a, 1.f, E);
    negcopy4_kernel<<<(int)(n1 / 4 + 255) / 256, 256, 0, stream>>>(
        (float4*)tx2a, (const float4*)x, n1 / 4);
    spmm_scatter_kernel<128><<<spmm_blocks, 256, 0, stream>>>(ei, nw, tx1a, tx2a, 2.f, E);

    // ---- Layer 1 fused GEMM + bias + ReLU ----
    dim3 g1((N + 127) / 128, CH / 16);
    cheb_gemm_kernel<128, 256><<<g1, 256, 0, stream>>>(x, tx1a, tx2a, W1, b1, h, N, 1);

    // ---- Layer 2 propagation ----
    long long n2 = (long long)N * CH;
    zero4_kernel<<<(int)(n2 / 4 + 255) / 256, 256, 0, stream>>>((float4*)tx1b, n2 / 4);
    spmm_scatter_kernel<256><<<spmm_blocks, 256, 0, stream>>>(ei, nw, h, tx1b, 1.f, E);
    negcopy4_kernel<<<(int)(n2 / 4 + 255) / 256, 256, 0, stream>>>(
        (float4*)tx2b, (const float4*)h, n2 / 4);
    spmm_scatter_kernel<256><<<spmm_blocks, 256, 0, stream>>>(ei, nw, tx1b, tx2b, 2.f, E);

    // ---- Layer 2 fused GEMM + bias ----
    dim3 g2((N + 127) / 128, CO / 16);
    cheb_gemm_kernel<256, 64><<<g2, 256, 0, stream>>>(h, tx1b, tx2b, W2, b2, out, N, 0);
}